// AdjustNodeArea_6356551598558
// MI455X (gfx1250) — compile-verified
//
#include <hip/hip_runtime.h>
#include <math.h>

// ---- Problem constants (from reference) ----
#define NUM_MOVABLE 1000000
#define NUM_FILLER  200000
#define NUM_NODES   1200000
#define NB          512              // RB_X == RB_Y == PB_X == PB_Y == 512
#define BSF         1.953125f        // (1000-0)/512, exact in fp32
#define KW          3
#define MIN_ROUTE   0.5f
#define MAX_ROUTE   2.0f
#define MIN_PIN     0.4f             // 1/2.5
#define MAX_PIN     2.5f
#define TOTAL_PLACE 1000000.0f
#define WS_BUDGET   50000.0f         // 0.1 * 500000
#define AREA_STOP   0.01f
#define ROUTE_STOP  0.01f
#define PIN_STOP    0.05f

constexpr int BLK          = 256;
constexpr int NBLOCKS_NODE = (NUM_NODES + BLK - 1) / BLK;   // 4688
constexpr int PAD          = ((NBLOCKS_NODE + 63) / 64) * 64; // 4736, multiple of 64 for WMMA reduce
constexpr int NSUM         = 5;

typedef float v2f __attribute__((ext_vector_type(2)));
typedef float v8f __attribute__((ext_vector_type(8)));

// ---------------- Kernel 0: zero the padded partial-sum region ----------------
__global__ void k_zero(float* __restrict__ p, int n) {
    int i = blockIdx.x * blockDim.x + threadIdx.x;
    if (i < n) p[i] = 0.0f;
}

// ---------------- Kernel 1: clamp utilization maps (once, L2-resident after) ----------------
__global__ void k_clamp_maps(const float* __restrict__ route, const float* __restrict__ pin,
                             float* __restrict__ routeC, float* __restrict__ pinC) {
    int i = blockIdx.x * blockDim.x + threadIdx.x;
    if (i < NB * NB) {
        float r  = route[i];
        float rc = r * r * sqrtf(r);                 // r^2.5 exactly
        routeC[i] = fminf(fmaxf(rc, MIN_ROUTE), MAX_ROUTE);
        pinC[i]   = fminf(fmaxf(pin[i], MIN_PIN), MAX_PIN);
    }
}

// ---------------- Kernel 2: per-node overlap pass + block partial sums ----------------
__global__ void __launch_bounds__(BLK)
k_node_pass(const float* __restrict__ pos,
            const float* __restrict__ nsx, const float* __restrict__ nsy,
            const float* __restrict__ routeC, const float* __restrict__ pinC,
            float* __restrict__ incArr, float* __restrict__ partials) {
    int i = blockIdx.x * blockDim.x + threadIdx.x;
    float v_old = 0.f, v_inc = 0.f, v_rex = 0.f, v_pex = 0.f, v_fil = 0.f;

    if (i < NUM_NODES) {
        float sx = nsx[i], sy = nsy[i];
        if (i < NUM_MOVABLE) {
            float px = pos[i], py = pos[NUM_NODES + i];
            float old_area = sx * sy;
            float xh = px + sx, yh = py + sy;
            int bx0 = (int)floorf(px / BSF);
            int by0 = (int)floorf(py / BSF);

            float ox[KW], oyv[KW];
            int   bxc[KW], byc[KW];
#pragma unroll
            for (int k = 0; k < KW; ++k) {
                int bx = bx0 + k;
                int cx = min(max(bx, 0), NB - 1);
                float blx = (float)cx * BSF;
                float o = fmaxf(fminf(xh, blx + BSF) - fmaxf(px, blx), 0.0f);
                ox[k]  = (bx >= 0 && bx < NB) ? o : 0.0f;
                bxc[k] = cx;

                int by = by0 + k;
                int cy = min(max(by, 0), NB - 1);
                float bly = (float)cy * BSF;
                float o2 = fmaxf(fminf(yh, bly + BSF) - fmaxf(py, bly), 0.0f);
                oyv[k] = (by >= 0 && by < NB) ? o2 : 0.0f;
                byc[k] = cy;
            }

            // Warm L2/L0 lines for the gather (gfx1250 global_prefetch_b8)
            __builtin_prefetch(routeC + bxc[0] * NB + byc[0], 0, 0);
            __builtin_prefetch(pinC   + bxc[0] * NB + byc[0], 0, 0);

            float r_area = 0.f, p_area = 0.f;
#pragma unroll
            for (int k = 0; k < KW; ++k) {
                const float* rrow = routeC + bxc[k] * NB;
                const float* prow = pinC   + bxc[k] * NB;
                float oxk = ox[k];
#pragma unroll
                for (int l = 0; l < KW; ++l) {
                    float w = oxk * oyv[l];
                    r_area = fmaf(w, rrow[byc[l]], r_area);
                    p_area = fmaf(w, prow[byc[l]], p_area);
                }
            }

            float inc = fmaxf(fmaxf(r_area, p_area) - old_area, 0.0f);
            incArr[i] = inc;
            v_old = old_area;
            v_inc = inc;
            v_rex = fmaxf(r_area - old_area, 0.0f);
            v_pex = fmaxf(p_area - old_area, 0.0f);
        } else {
            v_fil = sx * sy;   // filler old area
        }
    }

    // Deterministic block reduction of the 5 scalars
    __shared__ float sdata[BLK];
    float vals[NSUM] = {v_old, v_inc, v_rex, v_pex, v_fil};
#pragma unroll
    for (int q = 0; q < NSUM; ++q) {
        sdata[threadIdx.x] = vals[q];
        __syncthreads();
        for (int s = BLK / 2; s > 0; s >>= 1) {
            if (threadIdx.x < s) sdata[threadIdx.x] += sdata[threadIdx.x + s];
            __syncthreads();
        }
        if (threadIdx.x == 0) partials[q * PAD + blockIdx.x] = sdata[0];
        __syncthreads();
    }
}

// ---------------- Kernel 3: WMMA-based final reduction + scalar decision logic ----------------
// One workgroup of 5 full waves (160 threads); wave w reduces partial array w with
// V_WMMA_F32_16X16X4_F32: A = ones(16x4) => D[m][n] += sum_k B[k][n]. Since A is all-ones,
// the cross-lane sum of D's VGPR0 equals 2 * (sum of all 64 B elements) per issue,
// independent of the exact B lane layout.
__global__ void __launch_bounds__(160)
k_scalars(const float* __restrict__ partials, float* __restrict__ scalars) {
    int wave = threadIdx.x >> 5;
    int lane = threadIdx.x & 31;
    const float* p = partials + wave * PAD;

    v2f a; a.x = 1.0f; a.y = 1.0f;     // all-ones A (layout-independent)
    v8f c = {};
    for (int i = 0; i < PAD; i += 64) {
        v2f b;
        b.x = p[i + lane];
        b.y = p[i + 32 + lane];
        c = __builtin_amdgcn_wmma_f32_16x16x4_f32(false, a, false, b, (short)0, c, false, false);
    }
    float v = c[0];
#pragma unroll
    for (int off = 16; off >= 1; off >>= 1) v += __shfl_xor(v, off, 32);
    v *= 0.5f;                          // lanes 0-15 (M=0) and 16-31 (M=8) hold identical rows

    __shared__ float tot[NSUM];
    if (lane == 0) tot[wave] = v;
    __syncthreads();

    if (threadIdx.x == 0) {
        float old_sum = tot[0], inc_sum = tot[1], rex = tot[2], pex = tot[3], fil = tot[4];
        float budget  = fminf(WS_BUDGET, TOTAL_PLACE - old_sum);
        float raw     = budget / inc_sum;
        float scale   = fminf(fmaxf(raw, 0.0f), 1.0f);
        float s_app   = (raw <= 0.0f) ? 0.0f : scale;       // per-node applied scale
        float inc_eff = (raw <= 0.0f) ? 0.0f : inc_sum * scale;
        float new_sum = old_sum + inc_eff;
        float area_ratio = inc_eff / old_sum;
        bool rflag  = (rex / old_sum) > ROUTE_STOP;
        bool pflag  = (pex / old_sum) > PIN_STOP;
        bool adjust = (area_ratio > AREA_STOP) && (rflag || pflag);
        bool fcond  = adjust && ((new_sum + fil) > TOTAL_PLACE);
        float fratio = sqrtf(fmaxf(TOTAL_PLACE - new_sum, 0.0f) / fil);
        scalars[0] = adjust ? 1.0f : 0.0f;
        scalars[1] = s_app;
        scalars[2] = fcond ? 1.0f : 0.0f;
        scalars[3] = fratio;
    }
}

// ---------------- Kernel 4: elementwise finalize, writes all 4 outputs ----------------
__global__ void __launch_bounds__(BLK)
k_final(const float* __restrict__ pos,
        const float* __restrict__ nsx, const float* __restrict__ nsy,
        const float* __restrict__ incArr, const float* __restrict__ scalars,
        float* __restrict__ out) {
    int i = blockIdx.x * blockDim.x + threadIdx.x;
    if (i >= NUM_NODES) return;

    float sx = nsx[i], sy = nsy[i];
    float px = pos[i], py = pos[NUM_NODES + i];

    bool  adjust = scalars[0] != 0.0f;
    float s_app  = scalars[1];
    bool  fcond  = scalars[2] != 0.0f;
    float fratio = scalars[3];

    float nx = sx, ny = sy, qx = px, qy = py;
    if (i < NUM_MOVABLE) {
        if (adjust) {
            float old_area = sx * sy;
            float new_area = old_area + incArr[i] * s_app;   // s_app==0 reproduces raw<=0 path
            float m = sqrtf(new_area / old_area);
            nx = sx * m;  ny = sy * m;
            qx = px + 0.5f * (sx - nx);
            qy = py + 0.5f * (sy - ny);
        }
    } else {
        if (fcond) {
            nx = sx * fratio;  ny = sy * fratio;
            qx = px + 0.5f * (sx - nx);
            qy = py + 0.5f * (sy - ny);
        }
    }
    // new_pos = [x_m, x_f | y_m, y_f], then new_nsx, new_nsy (node order m..f matches i)
    out[i]                 = qx;
    out[NUM_NODES + i]     = qy;
    out[2 * NUM_NODES + i] = nx;
    out[3 * NUM_NODES + i] = ny;
}

extern "C" void kernel_launch(void* const* d_in, const int* in_sizes, int n_in,
                              void* d_out, int out_size, void* d_ws, size_t ws_size,
                              hipStream_t stream) {
    const float* pos   = (const float*)d_in[0];   // 2*NUM_NODES
    const float* nsx   = (const float*)d_in[1];   // NUM_NODES
    const float* nsy   = (const float*)d_in[2];   // NUM_NODES
    // d_in[3..5]: pin_offset_x/y, target_density — unused by reference math
    const float* route = (const float*)d_in[6];   // 512*512
    const float* pin   = (const float*)d_in[7];   // 512*512
    float* out = (float*)d_out;

    float* ws       = (float*)d_ws;
    float* routeC   = ws;                          // 262144
    float* pinC     = routeC + NB * NB;            // 262144
    float* incArr   = pinC + NB * NB;              // NUM_MOVABLE
    float* partials = incArr + NUM_MOVABLE;        // NSUM * PAD
    float* scalars  = partials + NSUM * PAD;       // 8

    k_zero<<<(NSUM * PAD + 255) / 256, 256, 0, stream>>>(partials, NSUM * PAD);
    k_clamp_maps<<<(NB * NB + 255) / 256, 256, 0, stream>>>(route, pin, routeC, pinC);
    k_node_pass<<<NBLOCKS_NODE, BLK, 0, stream>>>(pos, nsx, nsy, routeC, pinC, incArr, partials);
    k_scalars<<<1, 160, 0, stream>>>(partials, scalars);
    k_final<<<NBLOCKS_NODE, BLK, 0, stream>>>(pos, nsx, nsy, incArr, scalars, out);
}